// DeepLinearMemoryCell_6210522710454
// MI455X (gfx1250) — compile-verified
//
#include <hip/hip_runtime.h>
#include <hip/hip_bf16.h>

typedef float v2f __attribute__((ext_vector_type(2)));
typedef float v4f __attribute__((ext_vector_type(4)));
typedef float v8f __attribute__((ext_vector_type(8)));

#define B_   16
#define IN_  1024
#define H_   1024
#define DK_  1024
#define D1_  4096
#define LR_  0.01f

// ---------------------------------------------------------------------------
// Skinny GEMM:  C[16,N] = A[16,K] * op(B)
//   TRANSB=true : B stored [N,K] row-major  -> C = A * B^T
//   TRANSB=false: B stored [K,N] row-major  -> C = A * B
// One wave32 per 16x16 tile of C, K-loop of V_WMMA_F32_16X16X4_F32.
// ---------------------------------------------------------------------------
template <bool TRANSB>
__global__ __launch_bounds__(32)
void gemm16_wmma(const float* __restrict__ A, const float* __restrict__ Bm,
                 float* __restrict__ C, int K, int N)
{
    const int lane = threadIdx.x;
    const int half = lane >> 4;       // 0: K pair {0,1}, 1: K pair {2,3}
    const int r    = lane & 15;       // row of A frag / col of B frag
    const int n0   = blockIdx.x * 16;

    const float* Arow = A + (size_t)r * K;
    const float* Brow = TRANSB ? (Bm + (size_t)(n0 + r) * K) : nullptr;

    v8f acc = {};
    #pragma unroll 4
    for (int kb = 0; kb < K; kb += 4) {
        const int ka = kb + 2 * half;
        v2f a, b;
        a.x = Arow[ka];
        a.y = Arow[ka + 1];
        if (TRANSB) {
            b.x = Brow[ka];
            b.y = Brow[ka + 1];
        } else {
            b.x = Bm[(size_t)ka * N + n0 + r];
            b.y = Bm[(size_t)(ka + 1) * N + n0 + r];
        }
        // D = A(16x4) * B(4x16) + C   (fp32 WMMA, wave32)
        acc = __builtin_amdgcn_wmma_f32_16x16x4_f32(
            /*neg_a=*/false, a, /*neg_b=*/false, b,
            /*c_mod=*/(short)0, acc, /*reuse_a=*/false, /*reuse_b=*/false);
    }

    // C/D layout: VGPR v -> M = v + 8*half, N = lane&15
    #pragma unroll
    for (int v = 0; v < 8; ++v)
        C[(size_t)(v + 8 * half) * N + n0 + r] = acc[v];
}

// ---------------------------------------------------------------------------
// Elementwise helpers (exact GELU as in jax.nn.gelu(approximate=False))
// ---------------------------------------------------------------------------
__device__ __forceinline__ float gelu_f(float x) {
    return 0.5f * x * (1.0f + erff(x * 0.70710678118654752440f));
}
__device__ __forceinline__ float gelu_grad_f(float x) {
    const float inv_sqrt_2pi = 0.39894228040143267794f;
    float cdf = 0.5f * (1.0f + erff(x * 0.70710678118654752440f));
    float pdf = inv_sqrt_2pi * expf(-0.5f * x * x);
    return cdf + x * pdf;
}

__global__ __launch_bounds__(256)
void gelu_fwd_kernel(const float* __restrict__ z, float* __restrict__ g, int n) {
    int i = blockIdx.x * 256 + threadIdx.x;
    if (i < n) g[i] = gelu_f(z[i]);
}

__global__ __launch_bounds__(256)
void dh_kernel(const float* __restrict__ h, const float* __restrict__ v,
               float* __restrict__ dh, int n) {
    int i = blockIdx.x * 256 + threadIdx.x;
    if (i < n) dh[i] = (h[i] - v[i]) * (1.0f / (float)H_);
}

__global__ __launch_bounds__(256)
void dz_kernel(float* __restrict__ dg, const float* __restrict__ z, int n) {
    int i = blockIdx.x * 256 + threadIdx.x;
    if (i < n) dg[i] = dg[i] * gelu_grad_f(z[i]);
}

// ---------------------------------------------------------------------------
// Streaming rank-1 update (the 512 MB that dominates):
//   out[b,i,j] = W[i,j] - LR * rowv[b,i] * colv[b,j]
// v4f (ext-vector) loads; W stays L2-resident across the 16 batch passes.
// Non-temporal v4f stores so the 512 MB write stream doesn't thrash L2.
// NJ4LOG = log2(NJ/4); i is uniform per 256-thread block -> scalar row value.
// ---------------------------------------------------------------------------
__global__ __launch_bounds__(256)
void update_w_kernel(const float* __restrict__ W,
                     const float* __restrict__ rowv,   // [B, NI]
                     const float* __restrict__ colv,   // [B, NJ]
                     float* __restrict__ out,          // [B, NI, NJ]
                     int NI, int NJ, int NJ4LOG)
{
    const int b = blockIdx.y;
    const size_t t  = (size_t)blockIdx.x * 256 + threadIdx.x;  // over NI * NJ/4
    const int   i  = (int)(t >> NJ4LOG);
    const int   j4 = (int)(t & (((size_t)1 << NJ4LOG) - 1));

    const float s = -LR_ * rowv[(size_t)b * NI + i];
    const v4f wv = ((const v4f*)W)[((size_t)i << NJ4LOG) + j4];
    const v4f cv = ((const v4f*)(colv + (size_t)b * NJ))[j4];

    v4f o;
    o.x = fmaf(s, cv.x, wv.x);
    o.y = fmaf(s, cv.y, wv.y);
    o.z = fmaf(s, cv.z, wv.z);
    o.w = fmaf(s, cv.w, wv.w);

    v4f* dst = ((v4f*)out) + ((size_t)b * NI << NJ4LOG) + t;
    __builtin_nontemporal_store(o, dst);
}

// ---------------------------------------------------------------------------
// Driver
// ---------------------------------------------------------------------------
extern "C" void kernel_launch(void* const* d_in, const int* in_sizes, int n_in,
                              void* d_out, int out_size, void* d_ws, size_t ws_size,
                              hipStream_t stream)
{
    const float* x   = (const float*)d_in[0];  // [16,1024]
    const float* W_k = (const float*)d_in[1];  // [1024,1024]
    const float* W_v = (const float*)d_in[2];  // [1024,1024]
    const float* W_o = (const float*)d_in[3];  // [1024,1024]
    const float* W0  = (const float*)d_in[4];  // [4096,1024]
    const float* W1  = (const float*)d_in[5];  // [1024,4096]
    float* out = (float*)d_out;

    // workspace layout (floats), total 262144 floats = 1 MiB
    float* ws  = (float*)d_ws;
    float* k_t = ws;                 // 16*1024
    float* v_t = ws + 16384;         // 16*1024
    float* z   = ws + 32768;         // 16*4096
    float* g   = ws + 98304;         // 16*4096
    float* h   = ws + 163840;        // 16*1024
    float* dh  = ws + 180224;        // 16*1024
    float* dg  = ws + 196608;        // 16*4096 (becomes dz in place)

    const dim3 wave(32);

    // projections
    gemm16_wmma<true><<<DK_ / 16, wave, 0, stream>>>(x, W_k, k_t, IN_, DK_);
    gemm16_wmma<true><<<H_  / 16, wave, 0, stream>>>(x, W_v, v_t, IN_, H_);

    // fast-weight MLP forward
    gemm16_wmma<true><<<D1_ / 16, wave, 0, stream>>>(k_t, W0, z, DK_, D1_);
    gelu_fwd_kernel<<<(B_ * D1_) / 256, 256, 0, stream>>>(z, g, B_ * D1_);
    gemm16_wmma<true><<<H_ / 16, wave, 0, stream>>>(g, W1, h, D1_, H_);

    // backward through the MLP
    dh_kernel<<<(B_ * H_) / 256, 256, 0, stream>>>(h, v_t, dh, B_ * H_);
    gemm16_wmma<false><<<D1_ / 16, wave, 0, stream>>>(dh, W1, dg, H_, D1_);
    dz_kernel<<<(B_ * D1_) / 256, 256, 0, stream>>>(dg, z, B_ * D1_);

    // output projection -> d_out[0 : 16384]
    gemm16_wmma<true><<<IN_ / 16, wave, 0, stream>>>(h, W_o, out, H_, IN_);

    // weight updates (bandwidth-dominant): W0_new then W1_new, concat after output
    float* w0_out = out + (size_t)B_ * IN_;
    float* w1_out = w0_out + (size_t)B_ * D1_ * DK_;
    update_w_kernel<<<dim3((D1_ * (DK_ / 4)) / 256, B_), 256, 0, stream>>>(
        W0, dg, k_t, w0_out, D1_, DK_, 8);   // NJ/4 = 256 -> log2 = 8
    update_w_kernel<<<dim3((H_ * (D1_ / 4)) / 256, B_), 256, 0, stream>>>(
        W1, dh, g, w1_out, H_, D1_, 10);     // NJ/4 = 1024 -> log2 = 10
}